// OpenAD_PN2_61735859912971
// MI455X (gfx1250) — compile-verified
//
#include <hip/hip_runtime.h>
#include <math.h>

// ---------------------------------------------------------------------------
// OpenAD / PointNet++-MSG forward for MI455X (gfx1250, wave32).
// All dense layers (x @ W^T + b) run through V_WMMA_F32_16X16X4_F32 so the
// GEMM FLOPs (~76 GFLOP fp32 total) ride the CDNA5 matrix pipe at full fp32
// fidelity; irregular stages (FPS / ball query / kNN / softmax) are scalar
// kernels since they are latency- not FLOP-bound.
// ---------------------------------------------------------------------------

typedef __attribute__((ext_vector_type(2))) float v2f;
typedef __attribute__((ext_vector_type(8))) float v8f;

#define CDIV(a, b) (((a) + (b) - 1) / (b))

// ------------------------------ transpose (B,3,N) -> (B,N,3) ----------------
__global__ void k_xyz_to_pts(const float* __restrict__ xyz, float* __restrict__ pts,
                             int B, int N) {
  size_t i = (size_t)blockIdx.x * blockDim.x + threadIdx.x;
  size_t tot = (size_t)B * N * 3;
  if (i >= tot) return;
  int b = (int)(i / ((size_t)N * 3));
  int r = (int)(i % ((size_t)N * 3));
  int n = r / 3, c = r % 3;
  pts[i] = xyz[((size_t)b * 3 + c) * N + n];
}

// ------------------------------ farthest point sampling ---------------------
// Reproduces jax.lax.scan semantics: idx[j] = carry "far" at iteration entry.
__global__ void k_fps(const float* __restrict__ pts, int n, int np,
                      int* __restrict__ idx_out) {
  __shared__ float dist[4096];
  __shared__ float rv[256];
  __shared__ int   ri[256];
  int b = blockIdx.x;
  int t = threadIdx.x;
  const float* P = pts + (size_t)b * n * 3;
  for (int i = t; i < n; i += 256) dist[i] = 1e10f;
  int far = 0;
  __syncthreads();
  for (int j = 0; j < np; ++j) {
    if (t == 0) idx_out[(size_t)b * np + j] = far;
    float fx = P[(size_t)far * 3 + 0];
    float fy = P[(size_t)far * 3 + 1];
    float fz = P[(size_t)far * 3 + 2];
    float best = -1.0f; int bi = n - 1;
    for (int i = t; i < n; i += 256) {
      float dx = P[(size_t)i * 3 + 0] - fx;
      float dy = P[(size_t)i * 3 + 1] - fy;
      float dz = P[(size_t)i * 3 + 2] - fz;
      float d = dx * dx + dy * dy + dz * dz;
      float nd = fminf(dist[i], d);
      dist[i] = nd;
      if (nd > best || (nd == best && i < bi)) { best = nd; bi = i; }
    }
    rv[t] = best; ri[t] = bi;
    __syncthreads();
    for (int s = 128; s > 0; s >>= 1) {
      if (t < s) {
        if (rv[t + s] > rv[t] || (rv[t + s] == rv[t] && ri[t + s] < ri[t])) {
          rv[t] = rv[t + s]; ri[t] = ri[t + s];
        }
      }
      __syncthreads();
    }
    far = ri[0];
    __syncthreads();
  }
}

// ------------------------------ generic gather ------------------------------
__global__ void k_gather(const float* __restrict__ src, const int* __restrict__ idx,
                         float* __restrict__ dst, int B, int n, int s, int C) {
  size_t i = (size_t)blockIdx.x * blockDim.x + threadIdx.x;
  size_t tot = (size_t)B * s * C;
  if (i >= tot) return;
  int b = (int)(i / ((size_t)s * C));
  int r = (int)(i % ((size_t)s * C));
  int j = r / C, c = r % C;
  int id = idx[(size_t)b * s + j];
  dst[i] = src[((size_t)b * n + id) * C + c];
}

// ------------------------------ ball query ----------------------------------
// First k indices (ascending) with d <= r^2, pad with the first hit.
__global__ void k_ball(const float* __restrict__ pts, const float* __restrict__ nxyz,
                       int* __restrict__ out, int B, int n, int s, int k, float r2) {
  int q = blockIdx.x * blockDim.x + threadIdx.x;
  if (q >= B * s) return;
  int b = q / s;
  const float* P = pts + (size_t)b * n * 3;
  float qx = nxyz[(size_t)q * 3 + 0];
  float qy = nxyz[(size_t)q * 3 + 1];
  float qz = nxyz[(size_t)q * 3 + 2];
  int* o = out + (size_t)q * k;
  int cnt = 0;
  for (int i = 0; i < n && cnt < k; ++i) {
    float dx = P[(size_t)i * 3 + 0] - qx;
    float dy = P[(size_t)i * 3 + 1] - qy;
    float dz = P[(size_t)i * 3 + 2] - qz;
    if (dx * dx + dy * dy + dz * dz <= r2) o[cnt++] = i;
  }
  int f = (cnt > 0) ? o[0] : 0;
  for (; cnt < k; ++cnt) o[cnt] = f;
}

// --------------- group: gp[b,s,k,:] = [points[idx], xyz[idx]-new_xyz] -------
__global__ void k_group(const float* __restrict__ pts, const float* __restrict__ points,
                        const float* __restrict__ nxyz, const int* __restrict__ idx,
                        float* __restrict__ gp, int B, int n, int s, int k, int Cp) {
  size_t t = (size_t)blockIdx.x * blockDim.x + threadIdx.x;
  size_t tot = (size_t)B * s * k;
  if (t >= tot) return;
  int b = (int)(t / ((size_t)s * k));
  int r = (int)(t % ((size_t)s * k));
  int j = r / k;
  int id = idx[t];
  const float* ps = points + ((size_t)b * n + id) * Cp;
  float* g = gp + t * (Cp + 3);
  for (int c = 0; c < Cp; ++c) g[c] = ps[c];
  const float* pp = pts + ((size_t)b * n + id) * 3;
  const float* nc = nxyz + ((size_t)b * s + j) * 3;
  g[Cp + 0] = pp[0] - nc[0];
  g[Cp + 1] = pp[1] - nc[1];
  g[Cp + 2] = pp[2] - nc[2];
}

// ------------------------------ WMMA fp32 GEMM ------------------------------
// Y[M,N] = X[M,K] @ W[N,K]^T + bias[N].   One wave = 32x32 output tile
// (2x2 WMMA tiles, A/B fragments each reused twice per K-step).
//
// Fragment layouts (ISA 7.12.2, 16x16x4 f32):
//   A (16x4): lane half owns K pair {2*half, 2*half+1}, row = lane&15.
//   B (4x16): same K split, col = lane&15.
//   D: m = v + 8*half, n = lane&15.
//
// OOB rows/cols are CLAMPED, not masked: A-row r only feeds output row r and
// B-col c only feeds output col c, so clamped lanes produce garbage only in
// rows/cols that the store guards discard -> the K main loop has zero
// branching. Only the ragged K tail (K%4 != 0) needs true zero-masking,
// since OOB-K products would pollute valid outputs.
__global__ void k_gemm_wmma(const float* __restrict__ X, const float* __restrict__ W,
                            const float* __restrict__ bias, float* __restrict__ Y,
                            int M, int N, int K) {
  const int lane = threadIdx.x & 31;
  const int wave = threadIdx.x >> 5;
  const int m0 = blockIdx.x * 32;
  const int n0 = (blockIdx.y * 4 + wave) * 32;
  const int half = lane >> 4;
  const int t = lane & 15;
  const int kadj = half * 2;
  const int row0 = m0 + t, row1 = m0 + 16 + t;
  const int col0 = n0 + t, col1 = n0 + 16 + t;
  const bool r0v = row0 < M, r1v = row1 < M;
  const bool c0v = col0 < N, c1v = col1 < N;
  const float* x0 = X + (size_t)(r0v ? row0 : M - 1) * K + kadj;
  const float* x1 = X + (size_t)(r1v ? row1 : M - 1) * K + kadj;
  const float* w0 = W + (size_t)(c0v ? col0 : N - 1) * K + kadj;
  const float* w1 = W + (size_t)(c1v ? col1 : N - 1) * K + kadj;
  v8f acc00 = {}, acc01 = {}, acc10 = {}, acc11 = {};
  const int K4 = K & ~3;
  for (int k0 = 0; k0 < K4; k0 += 4) {
    if ((k0 & 31) == 0) {                       // scalar-uniform condition
      __builtin_prefetch(x0 + k0 + 128, 0, 0);  // -> global_prefetch_b8
      __builtin_prefetch(x1 + k0 + 128, 0, 0);
    }
    v2f a0, a1, b0, b1;
    a0[0] = x0[k0]; a0[1] = x0[k0 + 1];
    a1[0] = x1[k0]; a1[1] = x1[k0 + 1];
    b0[0] = w0[k0]; b0[1] = w0[k0 + 1];
    b1[0] = w1[k0]; b1[1] = w1[k0 + 1];
    acc00 = __builtin_amdgcn_wmma_f32_16x16x4_f32(false, a0, false, b0, (short)0, acc00, false, false);
    acc01 = __builtin_amdgcn_wmma_f32_16x16x4_f32(false, a0, false, b1, (short)0, acc01, false, false);
    acc10 = __builtin_amdgcn_wmma_f32_16x16x4_f32(false, a1, false, b0, (short)0, acc10, false, false);
    acc11 = __builtin_amdgcn_wmma_f32_16x16x4_f32(false, a1, false, b1, (short)0, acc11, false, false);
  }
  if (K4 < K) {   // ragged-K tail: zero-mask k >= K
    const bool e0 = (K4 + kadj) < K;
    const bool e1 = (K4 + kadj + 1) < K;
    v2f a0, a1, b0, b1;
    a0[0] = e0 ? x0[K4] : 0.0f; a0[1] = e1 ? x0[K4 + 1] : 0.0f;
    a1[0] = e0 ? x1[K4] : 0.0f; a1[1] = e1 ? x1[K4 + 1] : 0.0f;
    b0[0] = e0 ? w0[K4] : 0.0f; b0[1] = e1 ? w0[K4 + 1] : 0.0f;
    b1[0] = e0 ? w1[K4] : 0.0f; b1[1] = e1 ? w1[K4 + 1] : 0.0f;
    acc00 = __builtin_amdgcn_wmma_f32_16x16x4_f32(false, a0, false, b0, (short)0, acc00, false, false);
    acc01 = __builtin_amdgcn_wmma_f32_16x16x4_f32(false, a0, false, b1, (short)0, acc01, false, false);
    acc10 = __builtin_amdgcn_wmma_f32_16x16x4_f32(false, a1, false, b0, (short)0, acc10, false, false);
    acc11 = __builtin_amdgcn_wmma_f32_16x16x4_f32(false, a1, false, b1, (short)0, acc11, false, false);
  }
  const float bi0 = c0v ? bias[col0] : 0.0f;
  const float bi1 = c1v ? bias[col1] : 0.0f;
  for (int v = 0; v < 8; ++v) {
    int ma = m0 + v + 8 * half;
    int mb = ma + 16;
    if (ma < M) {
      if (c0v) Y[(size_t)ma * N + col0] = acc00[v] + bi0;
      if (c1v) Y[(size_t)ma * N + col1] = acc01[v] + bi1;
    }
    if (mb < M) {
      if (c0v) Y[(size_t)mb * N + col0] = acc10[v] + bi0;
      if (c1v) Y[(size_t)mb * N + col1] = acc11[v] + bi1;
    }
  }
}

// ------------------------------ per-channel mean/var ------------------------
__global__ void k_colstats(const float* __restrict__ Y, float* __restrict__ stats,
                           size_t M, int N) {
  __shared__ float s1[256], s2[256];
  int c = blockIdx.x;
  int t = threadIdx.x;
  float a = 0.0f, b = 0.0f;
  for (size_t m = t; m < M; m += 256) {
    float v = Y[m * N + c];
    a += v; b += v * v;
  }
  s1[t] = a; s2[t] = b;
  __syncthreads();
  for (int s = 128; s > 0; s >>= 1) {
    if (t < s) { s1[t] += s1[t + s]; s2[t] += s2[t + s]; }
    __syncthreads();
  }
  if (t == 0) {
    float mean = s1[0] / (float)M;
    stats[c] = mean;
    stats[N + c] = s2[0] / (float)M - mean * mean;
  }
}

__global__ void k_bn_act(float* __restrict__ Y, const float* __restrict__ stats,
                         const float* __restrict__ g, const float* __restrict__ be,
                         size_t M, int N, int relu) {
  size_t i = (size_t)blockIdx.x * blockDim.x + threadIdx.x;
  if (i >= M * (size_t)N) return;
  int c = (int)(i % N);
  float x = (Y[i] - stats[c]) * rsqrtf(stats[N + c] + 1e-5f) * g[c] + be[c];
  Y[i] = relu ? fmaxf(x, 0.0f) : x;
}

// --------------- max over the k axis of (BS, Kk, C) -> (BS, Ctot) -----------
__global__ void k_maxpool(const float* __restrict__ in, float* __restrict__ out,
                          int BS, int Kk, int C, int Ctot, int coff) {
  int i = blockIdx.x * blockDim.x + threadIdx.x;
  if (i >= BS * C) return;
  int r = i / C, c = i % C;
  const float* p = in + ((size_t)r * Kk) * C + c;
  float m = -3.4e38f;
  for (int k = 0; k < Kk; ++k) m = fmaxf(m, p[(size_t)k * C]);
  out[(size_t)r * Ctot + coff + c] = m;
}

// ------------------------------ column copy / broadcast ---------------------
__global__ void k_copy_cols(const float* __restrict__ src, float* __restrict__ dst,
                            size_t R, int Cs, int Cd, int off) {
  size_t i = (size_t)blockIdx.x * blockDim.x + threadIdx.x;
  if (i >= R * (size_t)Cs) return;
  size_t r = i / Cs;
  int c = (int)(i % Cs);
  dst[r * Cd + off + c] = src[i];
}

__global__ void k_bcast_cols(const float* __restrict__ src, float* __restrict__ dst,
                             int B, int n, int C, int Cd, int off) {
  size_t i = (size_t)blockIdx.x * blockDim.x + threadIdx.x;
  size_t tot = (size_t)B * n * C;
  if (i >= tot) return;
  int b = (int)(i / ((size_t)n * C));
  int r = (int)(i % ((size_t)n * C));
  int in = r / C, c = r % C;
  dst[((size_t)b * n + in) * Cd + off + c] = src[(size_t)b * C + c];
}

// ------------------- 3-NN inverse-distance interpolation --------------------
__global__ void k_interp3(const float* __restrict__ xyz1, const float* __restrict__ xyz2,
                          const float* __restrict__ p2, float* __restrict__ dst,
                          int B, int n1, int s2, int C2, int Cd, int off) {
  int q = blockIdx.x * blockDim.x + threadIdx.x;
  if (q >= B * n1) return;
  int b = q / n1;
  const float* x1 = xyz1 + (size_t)q * 3;
  const float* X2 = xyz2 + (size_t)b * s2 * 3;
  float bd[3] = {3e38f, 3e38f, 3e38f};
  int bi[3] = {0, 0, 0};
  float qx = x1[0], qy = x1[1], qz = x1[2];
  for (int j = 0; j < s2; ++j) {
    float dx = X2[(size_t)j * 3 + 0] - qx;
    float dy = X2[(size_t)j * 3 + 1] - qy;
    float dz = X2[(size_t)j * 3 + 2] - qz;
    float d = dx * dx + dy * dy + dz * dz;
    if (d < bd[2]) {                 // strict < keeps stable tie order
      int p = 2;
      while (p > 0 && d < bd[p - 1]) { bd[p] = bd[p - 1]; bi[p] = bi[p - 1]; --p; }
      bd[p] = d; bi[p] = j;
    }
  }
  float w0 = 1.0f / (bd[0] + 1e-8f);
  float w1 = 1.0f / (bd[1] + 1e-8f);
  float w2 = 1.0f / (bd[2] + 1e-8f);
  float ws = w0 + w1 + w2;
  w0 /= ws; w1 /= ws; w2 /= ws;
  const float* P2 = p2 + (size_t)b * s2 * C2;
  float* drow = dst + (size_t)q * Cd + off;
  for (int c = 0; c < C2; ++c)
    drow[c] = w0 * P2[(size_t)bi[0] * C2 + c] +
              w1 * P2[(size_t)bi[1] * C2 + c] +
              w2 * P2[(size_t)bi[2] * C2 + c];
}

// ------------------------------ bitonic sort of 1024 ints -------------------
__global__ void __launch_bounds__(1024) k_sort1024(int* __restrict__ idx, int m) {
  __shared__ int s[1024];
  int t = threadIdx.x;
  int* a = idx + (size_t)blockIdx.x * m;
  s[t] = a[t];
  __syncthreads();
  for (int k = 2; k <= 1024; k <<= 1) {
    for (int j = k >> 1; j > 0; j >>= 1) {
      int ixj = t ^ j;
      if (ixj > t) {
        bool up = ((t & k) == 0);
        int x = s[t], y = s[ixj];
        if ((x > y) == up) { s[t] = y; s[ixj] = x; }
      }
      __syncthreads();
    }
  }
  a[t] = s[t];
}

// -------- geometry relations: mean over 12-NN of [dxyz(3), dfeat(128)] ------
__global__ void k_georel(const float* __restrict__ pts, const float* __restrict__ feat,
                         const float* __restrict__ gpt, const float* __restrict__ gft,
                         float* __restrict__ out, int B, int n, int m) {
  int q = blockIdx.x * blockDim.x + threadIdx.x;
  if (q >= B * m) return;
  int b = q / m;
  const float* P = pts + (size_t)b * n * 3;
  const float* F = feat + (size_t)b * n * 128;
  float px = gpt[(size_t)q * 3 + 0];
  float py = gpt[(size_t)q * 3 + 1];
  float pz = gpt[(size_t)q * 3 + 2];
  float kd[12]; int ki[12];
  for (int j = 0; j < 12; ++j) { kd[j] = 3e38f; ki[j] = 0; }
  for (int i = 0; i < n; ++i) {
    float dx = P[(size_t)i * 3 + 0] - px;
    float dy = P[(size_t)i * 3 + 1] - py;
    float dz = P[(size_t)i * 3 + 2] - pz;
    float d = dx * dx + dy * dy + dz * dz;
    if (d < kd[11]) {
      int p = 11;
      while (p > 0 && d < kd[p - 1]) { kd[p] = kd[p - 1]; ki[p] = ki[p - 1]; --p; }
      kd[p] = d; ki[p] = i;
    }
  }
  float* o = out + (size_t)q * 131;
  for (int c = 0; c < 3; ++c) {
    float s = 0.0f;
    for (int j = 0; j < 12; ++j) s += P[(size_t)ki[j] * 3 + c];
    o[c] = s * (1.0f / 12.0f) - gpt[(size_t)q * 3 + c];
  }
  const float* fq = gft + (size_t)q * 128;
  for (int c = 0; c < 128; ++c) {
    float s = 0.0f;
    for (int j = 0; j < 12; ++j) s += F[(size_t)ki[j] * 128 + c];
    o[3 + c] = s * (1.0f / 12.0f) - fq[c];
  }
}

// ------------------------------ attention head ------------------------------
__global__ void k_attn(const float* __restrict__ xq, const float* __restrict__ text,
                       float* __restrict__ attn, int BN) {
  size_t i = (size_t)blockIdx.x * blockDim.x + threadIdx.x;
  if (i >= (size_t)BN * 19) return;
  size_t r = i / 19;
  int c = (int)(i % 19);
  const float* x = xq + r * 512;
  float s = 0.0f;
  for (int d = 0; d < 512; ++d) s += x[d] * text[(size_t)d * 19 + c];
  attn[i] = s;
}

__global__ void k_softmax19(float* __restrict__ attn, int BN) {
  int r = blockIdx.x * blockDim.x + threadIdx.x;
  if (r >= BN) return;
  float* a = attn + (size_t)r * 19;
  float mx = -3.4e38f;
  for (int c = 0; c < 19; ++c) mx = fmaxf(mx, a[c]);
  float s = 0.0f;
  for (int c = 0; c < 19; ++c) { float e = expf(a[c] - mx); a[c] = e; s += e; }
  float inv = 1.0f / s;
  for (int c = 0; c < 19; ++c) a[c] *= inv;
}

// softmax over the point axis (n) of attn*9, per (b, class)
__global__ void k_attn2(const float* __restrict__ attn, float* __restrict__ attn2,
                        int B, int n) {
  __shared__ float red[256];
  int b = blockIdx.x / 19, cls = blockIdx.x % 19;
  int t = threadIdx.x;
  const float* A = attn + (size_t)b * n * 19 + cls;
  float mx = -3.4e38f;
  for (int i = t; i < n; i += 256) mx = fmaxf(mx, A[(size_t)i * 19] * 9.0f);
  red[t] = mx; __syncthreads();
  for (int s = 128; s > 0; s >>= 1) {
    if (t < s) red[t] = fmaxf(red[t], red[t + s]);
    __syncthreads();
  }
  mx = red[0]; __syncthreads();
  float sm = 0.0f;
  for (int i = t; i < n; i += 256) sm += expf(A[(size_t)i * 19] * 9.0f - mx);
  red[t] = sm; __syncthreads();
  for (int s = 128; s > 0; s >>= 1) {
    if (t < s) red[t] += red[t + s];
    __syncthreads();
  }
  float inv = 1.0f / red[0];
  for (int i = t; i < n; i += 256)
    attn2[(size_t)b * n * 19 + (size_t)i * 19 + cls] =
        expf(A[(size_t)i * 19] * 9.0f - mx) * inv;
}

__global__ void k_rownorm(const float* __restrict__ X, float* __restrict__ out,
                          int R, int C) {
  int r = blockIdx.x * blockDim.x + threadIdx.x;
  if (r >= R) return;
  const float* x = X + (size_t)r * C;
  float s = 0.0f;
  for (int c = 0; c < C; ++c) s += x[c] * x[c];
  out[r] = sqrtf(s);
}

__global__ void k_wc(const float* __restrict__ xq, const float* __restrict__ attn2,
                     float* __restrict__ wc, int B, int n) {
  size_t i = (size_t)blockIdx.x * blockDim.x + threadIdx.x;
  if (i >= (size_t)B * 19 * 512) return;
  int b = (int)(i / (19 * 512));
  int r = (int)(i % (19 * 512));
  int c = r / 512, d = r % 512;
  float s = 0.0f;
  for (int j = 0; j < n; ++j)
    s += xq[((size_t)b * n + j) * 512 + d] * attn2[((size_t)b * n + j) * 19 + c];
  wc[i] = s;
}

__global__ void k_head(const float* __restrict__ xq, const float* __restrict__ wc,
                       const float* __restrict__ xqn, const float* __restrict__ wcn,
                       const float* __restrict__ scale_p, float* __restrict__ out,
                       int B, int n) {
  int r = blockIdx.x * blockDim.x + threadIdx.x;
  if (r >= B * n) return;
  int b = r / n, in = r % n;
  const float* x = xq + (size_t)r * 512;
  float scale = scale_p[0];
  float l[19];
  for (int c = 0; c < 19; ++c) {
    const float* w = wc + ((size_t)b * 19 + c) * 512;
    float s = 0.0f;
    for (int d = 0; d < 512; ++d) s += x[d] * w[d];
    l[c] = scale * s / (xqn[r] * wcn[b * 19 + c]);
  }
  float mx = -3.4e38f;
  for (int c = 0; c < 19; ++c) mx = fmaxf(mx, l[c]);
  float s = 0.0f;
  for (int c = 0; c < 19; ++c) s += expf(l[c] - mx);
  float lse = mx + logf(s);
  for (int c = 0; c < 19; ++c)
    out[((size_t)b * 19 + c) * n + in] = l[c] - lse;
}

__global__ void k_transpose_feat(const float* __restrict__ feat, float* __restrict__ out,
                                 int B, int N, int C) {
  size_t i = (size_t)blockIdx.x * blockDim.x + threadIdx.x;
  size_t tot = (size_t)B * N * C;
  if (i >= tot) return;
  int b = (int)(i / ((size_t)N * C));
  int r = (int)(i % ((size_t)N * C));
  int c = r / N, n = r % N;
  out[i] = feat[((size_t)b * N + n) * C + c];
}

// ---------------------------------------------------------------------------
// Host orchestration
// ---------------------------------------------------------------------------
namespace {

struct Layer { const float* w; const float* b; const float* g; const float* be; int cout; };

inline void gemm(const float* X, const float* W, const float* b, float* Y,
                 size_t M, int N, int K, hipStream_t st) {
  dim3 grid((unsigned)CDIV(M, 32), (unsigned)CDIV(N, 128));
  k_gemm_wmma<<<grid, 128, 0, st>>>(X, W, b, Y, (int)M, N, K);
}

inline const float* run_mlp(const Layer* L, int nL, const float* X, size_t M, int K0,
                            float* bufA, float* bufB, float* stats, hipStream_t st) {
  const float* cur = X;
  int K = K0;
  for (int li = 0; li < nL; ++li) {
    float* out = (li & 1) ? bufB : bufA;
    int N = L[li].cout;
    gemm(cur, L[li].w, L[li].b, out, M, N, K, st);
    k_colstats<<<N, 256, 0, st>>>(out, stats, M, N);
    size_t tot = M * (size_t)N;
    k_bn_act<<<(unsigned)CDIV(tot, 256), 256, 0, st>>>(out, stats, L[li].g, L[li].be, M, N, 1);
    cur = out;
    K = N;
  }
  return cur;
}

} // namespace

extern "C" void kernel_launch(void* const* d_in, const int* in_sizes, int n_in,
                              void* d_out, int out_size, void* d_ws, size_t ws_size,
                              hipStream_t stream) {
  (void)in_sizes; (void)n_in; (void)out_size; (void)ws_size;
  constexpr int B = 8, N0 = 4096, S1 = 512, S2 = 128;

  // ---- unpack inputs (setup_inputs() insertion order; layer order w,b,g,be)
  int cu = 0;
  auto FP = [&](void) { return (const float*)d_in[cu++]; };
  const float* xyz  = FP();
  const float* text = FP();
  auto POP = [&](int cout) {
    Layer L; L.w = FP(); L.b = FP(); L.g = FP(); L.be = FP(); L.cout = cout; return L;
  };
  Layer sa10[3] = { POP(32), POP(32), POP(64) };
  Layer sa11[3] = { POP(64), POP(64), POP(128) };
  Layer sa12[3] = { POP(64), POP(96), POP(128) };
  Layer sa20[3] = { POP(128), POP(128), POP(256) };
  Layer sa21[3] = { POP(128), POP(196), POP(256) };
  Layer sa3l[2] = { POP(256), POP(512) };
  Layer fp3l[1] = { POP(128) };
  Layer fp2l[1] = { POP(128) };
  Layer fp1l[1] = { POP(128) };
  const float* c1w  = FP();
  const float* c1b  = FP();
  const float* bn1g = FP();
  const float* bn1be = FP();
  const float* lscale = FP();

  // ---- workspace arena (~810 MB peak)
  float* wsp = (float*)d_ws;
  size_t off = 0;
  auto A = [&](size_t nf) { float* p = wsp + off; off += nf; return p; };
  float* pts       = A((size_t)B * N0 * 3);
  float* l1_xyz    = A((size_t)B * S1 * 3);
  float* l1_points = A((size_t)B * S1 * 320);
  float* l2_xyz    = A((size_t)B * S2 * 3);
  float* l2_points = A((size_t)B * S2 * 512);
  float* l3        = A((size_t)B * 512);
  float* l2f       = A((size_t)B * S2 * 128);
  float* l1f       = A((size_t)B * S1 * 128);
  float* l0_feat   = A((size_t)B * N0 * 128);
  float* xq        = A((size_t)B * N0 * 512);
  float* attn      = A((size_t)B * N0 * 19);
  float* attn2     = A((size_t)B * N0 * 19);
  float* wc        = A((size_t)B * 19 * 512);
  float* xqn       = A((size_t)B * N0);
  float* wcn       = A((size_t)B * 19);
  float* stats     = A(1024);
  int*   fidx1     = (int*)A((size_t)B * S1);
  int*   fidx2     = (int*)A((size_t)B * S2);
  int*   gidx      = (int*)A((size_t)B * 1024);
  float* gpt       = A((size_t)B * 1024 * 3);
  float* gft       = A((size_t)B * 1024 * 128);
  int*   bidx      = (int*)A((size_t)B * S1 * 128);
  float* gp        = A((size_t)131072 * 323);   // largest grouped tensor (sa2 br2)
  float* bufA      = A((size_t)524288 * 128);   // GEMM ping
  float* bufB      = A((size_t)524288 * 128);   // GEMM pong

  float* o0 = (float*)d_out;                       // log-softmax logits (B,19,N)
  float* o1 = o0 + (size_t)B * 19 * N0;            // pc_relations (B,1024,131)
  float* o2 = o1 + (size_t)B * 1024 * 131;         // global_feature (B,128,N)

  // ---- l0: (B,3,N) -> (B,N,3); l0_points == pts
  {
    size_t tot = (size_t)B * N0 * 3;
    k_xyz_to_pts<<<(unsigned)CDIV(tot, 256), 256, 0, stream>>>(xyz, pts, B, N0);
  }

  // ---- SA1 (multi-scale grouping on 4096 -> 512)
  k_fps<<<B, 256, 0, stream>>>(pts, N0, S1, fidx1);
  {
    size_t tot = (size_t)B * S1 * 3;
    k_gather<<<(unsigned)CDIV(tot, 256), 256, 0, stream>>>(pts, fidx1, l1_xyz, B, N0, S1, 3);
  }
  {
    const float radii[3] = {0.1f, 0.2f, 0.4f};
    const int   ks[3]    = {32, 64, 128};
    const Layer* mlps[3] = {sa10, sa11, sa12};
    int coff = 0;
    for (int br = 0; br < 3; ++br) {
      int k = ks[br];
      size_t rows = (size_t)B * S1 * k;
      k_ball<<<(unsigned)CDIV(B * S1, 256), 256, 0, stream>>>(
          pts, l1_xyz, bidx, B, N0, S1, k, radii[br] * radii[br]);
      k_group<<<(unsigned)CDIV(rows, 256), 256, 0, stream>>>(
          pts, pts, l1_xyz, bidx, gp, B, N0, S1, k, 3);
      const float* fin = run_mlp(mlps[br], 3, gp, rows, 6, bufA, bufB, stats, stream);
      int cout = mlps[br][2].cout;
      k_maxpool<<<(unsigned)CDIV(B * S1 * cout, 256), 256, 0, stream>>>(
          fin, l1_points, B * S1, k, cout, 320, coff);
      coff += cout;
    }
  }

  // ---- SA2 (512 -> 128)
  k_fps<<<B, 256, 0, stream>>>(l1_xyz, S1, S2, fidx2);
  {
    size_t tot = (size_t)B * S2 * 3;
    k_gather<<<(unsigned)CDIV(tot, 256), 256, 0, stream>>>(l1_xyz, fidx2, l2_xyz, B, S1, S2, 3);
  }
  {
    const float radii[2] = {0.4f, 0.8f};
    const int   ks[2]    = {64, 128};
    const Layer* mlps[2] = {sa20, sa21};
    int coff = 0;
    for (int br = 0; br < 2; ++br) {
      int k = ks[br];
      size_t rows = (size_t)B * S2 * k;
      k_ball<<<(unsigned)CDIV(B * S2, 256), 256, 0, stream>>>(
          l1_xyz, l2_xyz, bidx, B, S1, S2, k, radii[br] * radii[br]);
      k_group<<<(unsigned)CDIV(rows, 256), 256, 0, stream>>>(
          l1_xyz, l1_points, l2_xyz, bidx, gp, B, S1, S2, k, 320);
      const float* fin = run_mlp(mlps[br], 3, gp, rows, 323, bufA, bufB, stats, stream);
      int cout = mlps[br][2].cout;
      k_maxpool<<<(unsigned)CDIV(B * S2 * cout, 256), 256, 0, stream>>>(
          fin, l2_points, B * S2, k, cout, 512, coff);
      coff += cout;
    }
  }

  // ---- SA3 (global): [l2_xyz, l2_points] -> 515 -> 256 -> 512, max over 128
  {
    size_t R3 = (size_t)B * S2;
    k_copy_cols<<<(unsigned)CDIV(R3 * 3, 256), 256, 0, stream>>>(l2_xyz, gp, R3, 3, 515, 0);
    k_copy_cols<<<(unsigned)CDIV(R3 * 512, 256), 256, 0, stream>>>(l2_points, gp, R3, 512, 515, 3);
    const float* fin = run_mlp(sa3l, 2, gp, R3, 515, bufA, bufB, stats, stream);
    k_maxpool<<<(unsigned)CDIV(B * 512, 256), 256, 0, stream>>>(fin, l3, B, S2, 512, 512, 0);
  }

  // ---- FP3 (broadcast l3 to the 128 points): [l2_points(512), l3(512)] -> 128
  {
    size_t R3 = (size_t)B * S2;
    k_copy_cols<<<(unsigned)CDIV(R3 * 512, 256), 256, 0, stream>>>(l2_points, gp, R3, 512, 1024, 0);
    k_bcast_cols<<<(unsigned)CDIV(R3 * 512, 256), 256, 0, stream>>>(l3, gp, B, S2, 512, 1024, 512);
    const float* fin = run_mlp(fp3l, 1, gp, R3, 1024, bufA, bufB, stats, stream);
    hipMemcpyAsync(l2f, fin, R3 * 128 * sizeof(float), hipMemcpyDeviceToDevice, stream);
  }

  // ---- FP2: [l1_points(320), interp(l2f,128)] -> 128
  {
    size_t R2 = (size_t)B * S1;
    k_copy_cols<<<(unsigned)CDIV(R2 * 320, 256), 256, 0, stream>>>(l1_points, gp, R2, 320, 448, 0);
    k_interp3<<<(unsigned)CDIV(B * S1, 128), 128, 0, stream>>>(
        l1_xyz, l2_xyz, l2f, gp, B, S1, S2, 128, 448, 320);
    const float* fin = run_mlp(fp2l, 1, gp, R2, 448, bufA, bufB, stats, stream);
    hipMemcpyAsync(l1f, fin, R2 * 128 * sizeof(float), hipMemcpyDeviceToDevice, stream);
  }

  // ---- FP1: [pts(3), pts(3), interp(l1f,128)] -> 128 => l0_feat
  {
    size_t R1 = (size_t)B * N0;
    k_copy_cols<<<(unsigned)CDIV(R1 * 3, 256), 256, 0, stream>>>(pts, gp, R1, 3, 134, 0);
    k_copy_cols<<<(unsigned)CDIV(R1 * 3, 256), 256, 0, stream>>>(pts, gp, R1, 3, 134, 3);
    k_interp3<<<(unsigned)CDIV(B * N0, 128), 128, 0, stream>>>(
        pts, l1_xyz, l1f, gp, B, N0, S1, 128, 134, 6);
    const float* fin = run_mlp(fp1l, 1, gp, R1, 134, bufA, bufB, stats, stream);
    hipMemcpyAsync(l0_feat, fin, R1 * 128 * sizeof(float), hipMemcpyDeviceToDevice, stream);
  }

  // ---- geometry relations -> pc_relations (B,1024,131)
  k_fps<<<B, 256, 0, stream>>>(pts, N0, 1024, gidx);
  k_sort1024<<<B, 1024, 0, stream>>>(gidx, 1024);
  {
    size_t t3 = (size_t)B * 1024 * 3;
    size_t tf = (size_t)B * 1024 * 128;
    k_gather<<<(unsigned)CDIV(t3, 256), 256, 0, stream>>>(pts, gidx, gpt, B, N0, 1024, 3);
    k_gather<<<(unsigned)CDIV(tf, 256), 256, 0, stream>>>(l0_feat, gidx, gft, B, N0, 1024, 128);
    k_georel<<<(unsigned)CDIV(B * 1024, 64), 64, 0, stream>>>(
        pts, l0_feat, gpt, gft, o1, B, N0, 1024);
  }

  // ---- head: conv1 + bn1 -> xq; attention; cosine logits; log-softmax
  {
    size_t R1 = (size_t)B * N0;
    gemm(l0_feat, c1w, c1b, xq, R1, 512, 128, stream);
    k_colstats<<<512, 256, 0, stream>>>(xq, stats, R1, 512);
    k_bn_act<<<(unsigned)CDIV(R1 * 512, 256), 256, 0, stream>>>(
        xq, stats, bn1g, bn1be, R1, 512, 0);
    k_attn<<<(unsigned)CDIV(R1 * 19, 256), 256, 0, stream>>>(xq, text, attn, (int)R1);
    k_softmax19<<<(unsigned)CDIV(R1, 256), 256, 0, stream>>>(attn, (int)R1);
    k_attn2<<<B * 19, 256, 0, stream>>>(attn, attn2, B, N0);
    k_rownorm<<<(unsigned)CDIV(R1, 128), 128, 0, stream>>>(xq, xqn, (int)R1, 512);
    k_wc<<<(unsigned)CDIV((size_t)B * 19 * 512, 128), 128, 0, stream>>>(xq, attn2, wc, B, N0);
    k_rownorm<<<(unsigned)CDIV(B * 19, 64), 64, 0, stream>>>(wc, wcn, B * 19, 512);
    k_head<<<(unsigned)CDIV(R1, 128), 128, 0, stream>>>(xq, wc, xqn, wcn, lscale, o0, B, N0);
    size_t tg = (size_t)B * 128 * N0;
    k_transpose_feat<<<(unsigned)CDIV(tg, 256), 256, 0, stream>>>(l0_feat, o2, B, N0, 128);
  }
}